// RecurrentNetworkAttention_73160472920721
// MI455X (gfx1250) — compile-verified
//
#include <hip/hip_runtime.h>
#include <hip/hip_bf16.h>
#include <math.h>

typedef __attribute__((ext_vector_type(2))) float v2f;
typedef __attribute__((ext_vector_type(8))) float v8f;

#define DIMN   64
#define HEADS  4
#define BB     8
#define QQ     128
#define VV     128
#define NROWS  1024            // B*Q = B*V
#define MAT    (NROWS * DIMN)  // 65536 floats per (array, head)
#define ACC_OFF (4 * HEADS * MAT)  // after uq_re/uq_im/uv_re/uv_im blocks

// ---------------------------------------------------------------------------
// Phase A: 16 GEMMs  [1024 x 64] @ [64 x 64]  (heads x {re,im} x {q,v})
// One wave32 computes one 16x16 f32 tile via V_WMMA_F32_16X16X4_F32 (K=4/step).
// ws layout: [a = src*2+part][h][n][j], a: 0=q_re 1=q_im 2=v_re 3=v_im
// ---------------------------------------------------------------------------
__global__ void __launch_bounds__(256)
phaseA_gemm(const float* __restrict__ q, const float* __restrict__ vals,
            const float* __restrict__ Ure, const float* __restrict__ Uim,
            float* __restrict__ ws) {
    int gtid = blockIdx.x * blockDim.x + threadIdx.x;
    int wave = gtid >> 5;
    int lane = threadIdx.x & 31;

    int ntile = wave & 3;          // 4 N-tiles of 16 (Ncols = 64)
    int mtile = (wave >> 2) & 63;  // 64 M-tiles of 16 (Mrows = 1024)
    int var   = wave >> 8;         // 0..15
    int h     = var >> 2;
    int part  = (var >> 1) & 1;    // 0 = re, 1 = im
    int src   = var & 1;           // 0 = queries / U rows 0..63, 1 = values / U rows 64..127

    const float* A  = src ? vals : q;                        // [1024, 64]
    const float* Um = part ? Uim : Ure;                      // [H, 128, 64]
    const float* Bm = Um + (size_t)h * 128 * 64 + (size_t)src * 64 * 64;
    float* out = ws + ((size_t)(src * 2 + part) * HEADS + h) * MAT;

    int m0 = mtile * 16, n0 = ntile * 16;
    int half = lane >> 4;          // K sub-block: 0 -> K{0,1}, 1 -> K{2,3}
    int l15  = lane & 15;

    v8f acc = {};
    for (int k0 = 0; k0 < 64; k0 += 4) {
        int ka = k0 + half * 2;
        v2f a, b;
        a.x = A[(size_t)(m0 + l15) * 64 + ka];
        a.y = A[(size_t)(m0 + l15) * 64 + ka + 1];
        b.x = Bm[(size_t)(ka    ) * 64 + n0 + l15];
        b.y = Bm[(size_t)(ka + 1) * 64 + n0 + l15];
        acc = __builtin_amdgcn_wmma_f32_16x16x4_f32(false, a, false, b,
                                                    (short)0, acc, false, false);
    }
    // C/D layout: VGPR r -> row m0 + r + half*8, col n0 + l15
    for (int r = 0; r < 8; ++r)
        out[(size_t)(m0 + r + half * 8) * 64 + n0 + l15] = acc[r];
}

// ---------------------------------------------------------------------------
// Phase B: EUNN recurrence, 128 steps. One wave32 per (head, n) state row.
// Lane L owns complex elements (2L, 2L+1). Layer-1 rotations intra-lane;
// layer-2 rotations via +-1 lane shuffles. All trig hoisted out of the loop.
// ---------------------------------------------------------------------------
__global__ void __launch_bounds__(256)
recurrence(float* __restrict__ ws,
           const float* __restrict__ bias,
           const float* __restrict__ theta1, const float* __restrict__ phi1,
           const float* __restrict__ theta2, const float* __restrict__ phi2,
           const float* __restrict__ omega) {
    const float* uq_re = ws + (size_t)0 * HEADS * MAT;
    const float* uq_im = ws + (size_t)1 * HEADS * MAT;
    const float* uv_re = ws + (size_t)2 * HEADS * MAT;
    const float* uv_im = ws + (size_t)3 * HEADS * MAT;
    float* acc = ws + ACC_OFF;   // [1024][256]

    int gtid = blockIdx.x * blockDim.x + threadIdx.x;
    int wave = gtid >> 5;
    int lane = threadIdx.x & 31;
    int h = wave >> 10;          // head
    int n = wave & 1023;         // row in [0, 1024)
    int b = n >> 7;              // batch = n / Q

    // ---- hoisted per-lane constants ----
    float t1 = theta1[h * 32 + lane], p1 = phi1[h * 32 + lane];
    float c1 = cosf(t1), s1 = sinf(t1);
    float e1r = cosf(p1), e1i = sinf(p1);

    float c2s = 1.f, s2s = 0.f, e2r = 1.f, e2i = 0.f;   // pair (2L+1, 2L+2)
    if (lane < 31) {
        float t2 = theta2[h * 31 + lane], p2 = phi2[h * 31 + lane];
        c2s = cosf(t2); s2s = sinf(t2); e2r = cosf(p2); e2i = sinf(p2);
    }
    float c2p = 1.f, s2p = 0.f;                          // pair (2L-1, 2L)
    if (lane > 0) {
        float t2 = theta2[h * 31 + lane - 1];
        c2p = cosf(t2); s2p = sinf(t2);
    }
    float om0 = omega[h * 64 + 2 * lane], om1 = omega[h * 64 + 2 * lane + 1];
    float omr0 = cosf(om0), omi0 = sinf(om0);
    float omr1 = cosf(om1), omi1 = sinf(om1);
    float b0 = bias[h * 64 + 2 * lane], b1 = bias[h * 64 + 2 * lane + 1];

    size_t qoff = ((size_t)h * NROWS + n) * 64 + 2 * lane;
    float uq0r = uq_re[qoff],     uq0i = uq_im[qoff];
    float uq1r = uq_re[qoff + 1], uq1i = uq_im[qoff + 1];

    float h0r = 0.f, h0i = 0.f, h1r = 0.f, h1i = 0.f;
    size_t vbase = ((size_t)h * NROWS + (size_t)b * VV) * 64 + 2 * lane;

    for (int step = 0; step < VV; ++step) {
        size_t vo = vbase + (size_t)step * 64;
        float uv0r = uv_re[vo],     uv0i = uv_im[vo];
        float uv1r = uv_re[vo + 1], uv1i = uv_im[vo + 1];

        // layer 1: pair (2L, 2L+1), intra-lane
        float tr  = c1 * h0r - s1 * h1r, ti = c1 * h0i - s1 * h1i;
        float g0r = e1r * tr - e1i * ti, g0i = e1r * ti + e1i * tr;
        float g1r = s1 * h0r + c1 * h1r, g1i = s1 * h0i + c1 * h1i;

        // layer 2: pair (2L+1, 2L+2) -> neighbor exchange
        float pg1r = __shfl_up(g1r, 1, 32);
        float pg1i = __shfl_up(g1i, 1, 32);
        float ng0r = __shfl_down(g0r, 1, 32);
        float ng0i = __shfl_down(g0i, 1, 32);

        float f0r = (lane == 0) ? g0r : (s2p * pg1r + c2p * g0r);
        float f0i = (lane == 0) ? g0i : (s2p * pg1i + c2p * g0i);
        float t2r = c2s * g1r - s2s * ng0r, t2i = c2s * g1i - s2s * ng0i;
        float f1r = (lane == 31) ? g1r : (e2r * t2r - e2i * t2i);
        float f1i = (lane == 31) ? g1i : (e2r * t2i + e2i * t2r);

        // diagonal phase exp(i*omega), then + u  (u = Uq + Uv)
        float z0r = omr0 * f0r - omi0 * f0i + uq0r + uv0r;
        float z0i = omr0 * f0i + omi0 * f0r + uq0i + uv0i;
        float z1r = omr1 * f1r - omi1 * f1i + uq1r + uv1r;
        float z1i = omr1 * f1i + omi1 * f1r + uq1i + uv1i;

        // modReLU
        float m0 = sqrtf(z0r * z0r + z0i * z0i);
        float sc0 = fmaxf(m0 + b0, 0.f) / (m0 + 1e-5f);
        h0r = z0r * sc0; h0i = z0i * sc0;
        float m1 = sqrtf(z1r * z1r + z1i * z1i);
        float sc1 = fmaxf(m1 + b1, 0.f) / (m1 + 1e-5f);
        h1r = z1r * sc1; h1i = z1i * sc1;
    }

    acc[(size_t)n * (HEADS * DIMN) + h * DIMN + 2 * lane]     = h0r;
    acc[(size_t)n * (HEADS * DIMN) + h * DIMN + 2 * lane + 1] = h1r;
}

// ---------------------------------------------------------------------------
// Phase C: y = acc[1024,256] @ W[256,64] + b  -> d_out. WMMA f32, K=256.
// ---------------------------------------------------------------------------
__global__ void __launch_bounds__(256)
dense_out(const float* __restrict__ ws, const float* __restrict__ W,
          const float* __restrict__ bvec, float* __restrict__ out) {
    const float* acc = ws + ACC_OFF;
    int gtid = blockIdx.x * blockDim.x + threadIdx.x;
    int wave = gtid >> 5;
    int lane = threadIdx.x & 31;
    int ntile = wave & 3;
    int mtile = wave >> 2;       // 0..63
    int half = lane >> 4, l15 = lane & 15;
    int m0 = mtile * 16, n0 = ntile * 16;

    v8f c = {};
    for (int k0 = 0; k0 < 256; k0 += 4) {
        int ka = k0 + half * 2;
        v2f a, b;
        a.x = acc[(size_t)(m0 + l15) * 256 + ka];
        a.y = acc[(size_t)(m0 + l15) * 256 + ka + 1];
        b.x = W[(size_t)(ka    ) * 64 + n0 + l15];
        b.y = W[(size_t)(ka + 1) * 64 + n0 + l15];
        c = __builtin_amdgcn_wmma_f32_16x16x4_f32(false, a, false, b,
                                                  (short)0, c, false, false);
    }
    float badd = bvec[n0 + l15];
    for (int r = 0; r < 8; ++r)
        out[(size_t)(m0 + r + half * 8) * 64 + n0 + l15] = c[r] + badd;
}

extern "C" void kernel_launch(void* const* d_in, const int* in_sizes, int n_in,
                              void* d_out, int out_size, void* d_ws, size_t ws_size,
                              hipStream_t stream) {
    const float* q    = (const float*)d_in[0];
    const float* vals = (const float*)d_in[1];
    const float* Ure  = (const float*)d_in[2];
    const float* Uim  = (const float*)d_in[3];
    const float* bias = (const float*)d_in[4];
    const float* th1  = (const float*)d_in[5];
    const float* ph1  = (const float*)d_in[6];
    const float* th2  = (const float*)d_in[7];
    const float* ph2  = (const float*)d_in[8];
    const float* om   = (const float*)d_in[9];
    const float* W    = (const float*)d_in[10];
    const float* bd   = (const float*)d_in[11];
    float* ws  = (float*)d_ws;
    float* out = (float*)d_out;

    // Phase A: 16 variants * 64 Mtiles * 4 Ntiles = 4096 waves
    phaseA_gemm<<<512, 256, 0, stream>>>(q, vals, Ure, Uim, ws);
    // Phase B: H * N = 4096 waves
    recurrence<<<512, 256, 0, stream>>>(ws, bias, th1, ph1, th2, ph2, om);
    // Phase C: 64 * 4 = 256 waves
    dense_out<<<32, 256, 0, stream>>>(ws, W, bd, out);
}